// Model_72919954752197
// MI455X (gfx1250) — compile-verified
//
#include <hip/hip_runtime.h>
#include <math.h>
#include <stdint.h>

// ---------------------------------------------------------------------------
// Problem constants (match reference)
// ---------------------------------------------------------------------------
#define N1   60000   // NUM_NODES_G1
#define N2   50000   // NUM_NODES_G2
#define NREL 16
#define NB   4
#define DIM  128
#define LCOM 100

typedef __attribute__((ext_vector_type(2))) float v2f;
typedef __attribute__((ext_vector_type(8))) float v8f;

// ---------------------------------------------------------------------------
// Utility kernels
// ---------------------------------------------------------------------------
__global__ void zero_f32(float* __restrict__ p, size_t n) {
    size_t i = (size_t)blockIdx.x * blockDim.x + threadIdx.x;
    size_t stride = (size_t)gridDim.x * blockDim.x;
    for (; i < n; i += stride) p[i] = 0.0f;
}

__global__ void count_deg(const int* __restrict__ dst, float* __restrict__ deg, int e) {
    int i = blockIdx.x * blockDim.x + threadIdx.x;
    if (i < e) atomicAdd(&deg[dst[i]], 1.0f);
}

__global__ void count_rel(const int* __restrict__ dst, const int* __restrict__ et,
                          float* __restrict__ cnt, int e) {
    int i = blockIdx.x * blockDim.x + threadIdx.x;
    if (i < e) atomicAdd(&cnt[dst[i] * NREL + et[i]], 1.0f);
}

// Concept layer: agg[dst] += x[src]   (one wave per edge, float4 per lane)
__global__ __launch_bounds__(256) void concept_scatter(
    const float* __restrict__ x, const int* __restrict__ src,
    const int* __restrict__ dst, float* __restrict__ agg, int e) {
    int gid  = blockIdx.x * blockDim.x + threadIdx.x;
    int eid  = gid >> 5;
    int lane = gid & 31;
    if (eid >= e) return;
    int s = src[eid], d = dst[eid];
    const float4 v = *(const float4*)(x + (size_t)s * DIM + lane * 4);
    float* o = agg + (size_t)d * DIM + lane * 4;
    atomicAdd(o + 0, v.x);
    atomicAdd(o + 1, v.y);
    atomicAdd(o + 2, v.z);
    atomicAdd(o + 3, v.w);
}

// x_g1 = relu(x + agg / max(deg,1))
__global__ void concept_finalize(const float* __restrict__ x, const float* __restrict__ agg,
                                 const float* __restrict__ deg, float* __restrict__ xg1,
                                 size_t n) {
    size_t i = (size_t)blockIdx.x * blockDim.x + threadIdx.x;
    size_t stride = (size_t)gridDim.x * blockDim.x;
    for (; i < n; i += stride) {
        float dgr = fmaxf(deg[i >> 7], 1.0f);
        xg1[i] = fmaxf(x[i] + agg[i] / dgr, 0.0f);
    }
}

// RGCN edge scatter: for b in 0..3:  aggb[b][dst] += comp[t][b]/max(cnt,1) * x[src]
__global__ __launch_bounds__(256) void rgcn_edge_scatter(
    const float* __restrict__ x, const int* __restrict__ src, const int* __restrict__ dst,
    const int* __restrict__ et, const float* __restrict__ comp, const float* __restrict__ cnt,
    float* __restrict__ aggb, int e) {
    int gid  = blockIdx.x * blockDim.x + threadIdx.x;
    int eid  = gid >> 5;
    int lane = gid & 31;
    if (eid >= e) return;
    int s = src[eid], d = dst[eid], t = et[eid];
    float norm = 1.0f / fmaxf(cnt[d * NREL + t], 1.0f);
    const float4 v = *(const float4*)(x + (size_t)s * DIM + lane * 4);
#pragma unroll
    for (int b = 0; b < NB; ++b) {
        float c = comp[t * NB + b] * norm;
        float* o = aggb + ((size_t)b * N2 + d) * DIM + lane * 4;
        atomicAdd(o + 0, c * v.x);
        atomicAdd(o + 1, c * v.y);
        atomicAdd(o + 2, c * v.z);
        atomicAdd(o + 3, c * v.w);
    }
}

// ---------------------------------------------------------------------------
// Weight repack: W[k][n] -> fragment order Wp[mat][s=k/4][t=n/16][lane][2]
// with lane = (half<<4)|(n&15), half = (k&3)>>1, comp = k&1.
// A wave then reads its B fragment as one contiguous v2f per lane.
// ---------------------------------------------------------------------------
__global__ void pack_weights(const float* __restrict__ root, const float* __restrict__ basis,
                             float* __restrict__ Wp) {
    int i = blockIdx.x * blockDim.x + threadIdx.x;
    if (i >= 5 * DIM * DIM) return;
    int mat = i >> 14;
    int rem = i & 16383;
    int k = rem >> 7, n = rem & 127;
    const float* W = (mat == 0) ? root : (basis + (size_t)(mat - 1) * DIM * DIM);
    int s = k >> 2, r = k & 3;
    int half = r >> 1, comp = r & 1;
    int t = n >> 4;
    int lane = (half << 4) | (n & 15);
    Wp[((((size_t)mat * 32 + s) * 8 + t) * 32 + lane) * 2 + comp] = W[(size_t)k * DIM + n];
}

// ---------------------------------------------------------------------------
// Async global->LDS tile stage (CDNA5): 16 rows x 128 floats, b128 chunks.
// LDS row stride 132 floats: 16B-aligned rows, bank = (4*row + k) % 64 -> no
// conflicts across the 16 lanes of a fragment read.
// ---------------------------------------------------------------------------
#define AROW 132

__device__ __forceinline__ void async_tile_load(const float* __restrict__ g,
                                                float* ldsbuf, int tid) {
    for (int i = tid; i < 512; i += 256) {          // 512 x 16B chunks
        int r = i >> 5, c4 = i & 31;
        unsigned      lds = (unsigned)(uintptr_t)(ldsbuf + r * AROW + c4 * 4);
        uint64_t      ga  = (uint64_t)(uintptr_t)(g + (size_t)r * DIM + c4 * 4);
        asm volatile("global_load_async_to_lds_b128 %0, %1, off"
                     :: "v"(lds), "v"(ga) : "memory");
    }
}

__device__ __forceinline__ const float* tile_src(const float* x, const float* aggb,
                                                 int mat, int m0) {
    return (mat == 0) ? (x + (size_t)m0 * DIM)
                      : (aggb + ((size_t)(mat - 1) * N2 + m0) * DIM);
}

// ---------------------------------------------------------------------------
// Fused RGCN GEMM:  out = x@root + bias + sum_b aggb[b]@basis[b]  (+opt relu)
// fp32 WMMA 16x16x4. Block = 256 thr = 8 waves; M-tile 16, wave w -> N-tile w.
// A tiles double-buffered in LDS via async loads; next tile prefetched while
// the current tile's 32 WMMAs run. B fragments from the packed weight buffer.
// ---------------------------------------------------------------------------
__global__ __launch_bounds__(256) void rgcn_gemm_wmma(
    const float* __restrict__ x, const float* __restrict__ aggb,
    const float* __restrict__ Wp, const float* __restrict__ bias,
    float* __restrict__ out, int relu_out) {
    __shared__ float As[2][16 * AROW];
    const int tid  = threadIdx.x;
    const int wave = tid >> 5;
    const int lane = tid & 31;
    const int half = lane >> 4;
    const int l16  = lane & 15;
    const int m0   = blockIdx.x * 16;

    async_tile_load(tile_src(x, aggb, 0, m0), As[0], tid);

    v8f acc = {};
    for (int mat = 0; mat < 5; ++mat) {
        asm volatile("s_wait_asynccnt 0x0" ::: "memory");
        __syncthreads();   // tile `mat` fully in LDS; all waves done with buf^1
        if (mat + 1 < 5)
            async_tile_load(tile_src(x, aggb, mat + 1, m0), As[(mat + 1) & 1], tid);
        const float* Abuf = As[mat & 1];
        const float* Bmat = Wp + (size_t)mat * (DIM * DIM);
#pragma unroll 4
        for (int s = 0; s < 32; ++s) {
            int ka = 4 * s + 2 * half;
            v2f a = *(const v2f*)(Abuf + l16 * AROW + ka);
            v2f b = *(const v2f*)(Bmat + (((size_t)s * 8 + wave) * 32 + lane) * 2);
            acc = __builtin_amdgcn_wmma_f32_16x16x4_f32(
                false, a, false, b, (short)0, acc, false, false);
        }
    }
    float bn = bias[wave * 16 + l16];
#pragma unroll
    for (int r = 0; r < 8; ++r) {
        int m = m0 + r + 8 * half;       // C/D layout: lanes>=16 hold M=8..15
        float v = acc[r] + bn;
        if (relu_out) v = fmaxf(v, 0.0f);
        out[(size_t)m * DIM + wave * 16 + l16] = v;
    }
}

// tyw[l][d] = x_g1[left_common[l]][d] * w[d]
__global__ void make_tyw(const float* __restrict__ xg1, const int* __restrict__ lc,
                         const float* __restrict__ w, float* __restrict__ tyw) {
    int i = blockIdx.x * blockDim.x + threadIdx.x;
    if (i < LCOM * DIM) {
        int l = i >> 7, d = i & 127;
        tyw[i] = xg1[(size_t)lc[l] * DIM + d] * w[d];
    }
}

// ---------------------------------------------------------------------------
// Logits GEMM:  logits[n][l] = h2[n][:] . tyw[l][:]   (N padded 100 -> 112)
// tyw staged via async b128 loads into padded LDS; pad rows zero-filled.
// ---------------------------------------------------------------------------
__global__ __launch_bounds__(256) void logits_wmma(
    const float* __restrict__ h2, const float* __restrict__ tyw,
    float* __restrict__ logits) {
    __shared__ float Bs[112 * AROW];
    const int tid  = threadIdx.x;
    const int wave = tid >> 5;
    const int lane = tid & 31;
    const int half = lane >> 4;
    const int l16  = lane & 15;
    const int m0   = blockIdx.x * 16;

    for (int i = tid; i < 12 * AROW; i += 256)       // zero pad rows 100..111
        Bs[100 * AROW + i] = 0.0f;
    for (int i = tid; i < LCOM * 32; i += 256) {     // async-stage 100 rows
        int r = i >> 5, c4 = i & 31;
        unsigned lds = (unsigned)(uintptr_t)(&Bs[r * AROW + c4 * 4]);
        uint64_t ga  = (uint64_t)(uintptr_t)(tyw + (size_t)r * DIM + c4 * 4);
        asm volatile("global_load_async_to_lds_b128 %0, %1, off"
                     :: "v"(lds), "v"(ga) : "memory");
    }
    asm volatile("s_wait_asynccnt 0x0" ::: "memory");
    __syncthreads();

    if (wave < 7) {                       // wave-uniform: EXEC all-ones inside
        const int n0 = wave * 16;
        v8f acc = {};
#pragma unroll 4
        for (int k0 = 0; k0 < DIM; k0 += 4) {
            int ka = k0 + 2 * half;
            v2f a = *(const v2f*)(h2 + (size_t)(m0 + l16) * DIM + ka);
            v2f b = *(const v2f*)(&Bs[(n0 + l16) * AROW + ka]);
            acc = __builtin_amdgcn_wmma_f32_16x16x4_f32(
                false, a, false, b, (short)0, acc, false, false);
        }
        int n = n0 + l16;
        if (n < LCOM) {
#pragma unroll
            for (int r = 0; r < 8; ++r) {
                int m = m0 + r + 8 * half;
                logits[(size_t)m * LCOM + n] = acc[r];
            }
        }
    }
}

// Row softmax, one wave32 per row of 100
__global__ __launch_bounds__(256) void softmax_rows(float* __restrict__ out, int nrows) {
    int row  = blockIdx.x * 8 + (threadIdx.x >> 5);
    int lane = threadIdx.x & 31;
    if (row >= nrows) return;
    float* p = out + (size_t)row * LCOM;
    float m = -INFINITY;
    for (int c = lane; c < LCOM; c += 32) m = fmaxf(m, p[c]);
#pragma unroll
    for (int off = 16; off; off >>= 1) m = fmaxf(m, __shfl_xor(m, off, 32));
    float s = 0.0f;
    for (int c = lane; c < LCOM; c += 32) s += expf(p[c] - m);
#pragma unroll
    for (int off = 16; off; off >>= 1) s += __shfl_xor(s, off, 32);
    float inv = 1.0f / s;
    for (int c = lane; c < LCOM; c += 32) p[c] = expf(p[c] - m) * inv;
}

// ---------------------------------------------------------------------------
// Host launcher
// ---------------------------------------------------------------------------
static inline int cdiv(long a, long b) { return (int)((a + b - 1) / b); }

extern "C" void kernel_launch(void* const* d_in, const int* in_sizes, int n_in,
                              void* d_out, int out_size, void* d_ws, size_t ws_size,
                              hipStream_t stream) {
    const int*   ei_g2  = (const int*)d_in[0];   // (2, E2)
    const int*   et_g2  = (const int*)d_in[1];   // (E2,)
    const int*   ei_g1  = (const int*)d_in[2];   // (2, E1)
    const int*   lc     = (const int*)d_in[3];   // (100,)
    const float* x_in   = (const float*)d_in[4]; // (60000,128)
    const float* basis1 = (const float*)d_in[5];
    const float* comp1  = (const float*)d_in[6];
    const float* root1  = (const float*)d_in[7];
    const float* bias1  = (const float*)d_in[8];
    const float* basis2 = (const float*)d_in[9];
    const float* comp2  = (const float*)d_in[10];
    const float* root2  = (const float*)d_in[11];
    const float* bias2  = (const float*)d_in[12];
    const float* wvec   = (const float*)d_in[13];

    const int E2 = in_sizes[0] / 2;
    const int E1 = in_sizes[2] / 2;
    const int* src_g2 = ei_g2;        const int* dst_g2 = ei_g2 + E2;
    const int* src_g1 = ei_g1;        const int* dst_g1 = ei_g1 + E1;

    // workspace layout (floats)
    float* ws   = (float*)d_ws;
    float* xg1  = ws;                                  // N1*DIM
    float* aggA = xg1  + (size_t)N1 * DIM;             // N1*DIM (concept agg; reused as h2)
    float* deg  = aggA + (size_t)N1 * DIM;             // N1
    float* cnt  = deg  + N1;                           // N2*NREL
    float* aggb = cnt  + (size_t)N2 * NREL;            // NB*N2*DIM
    float* h1   = aggb + (size_t)NB * N2 * DIM;        // N2*DIM
    float* tyw  = h1   + (size_t)N2 * DIM;             // LCOM*DIM
    float* Wp1  = tyw  + (size_t)LCOM * DIM;           // 5*DIM*DIM packed weights L1
    float* Wp2  = Wp1  + (size_t)5 * DIM * DIM;        // 5*DIM*DIM packed weights L2
    float* h2   = aggA;                                // reuse after concept layer

    float* logits = (float*)d_out;                     // (N2, LCOM)

    const int ZB = 2048;  // blocks for grid-stride zeroing

    // ---- Weight fragment repack (independent of graph phases) ----
    pack_weights<<<cdiv(5 * DIM * DIM, 256), 256, 0, stream>>>(root1, basis1, Wp1);
    pack_weights<<<cdiv(5 * DIM * DIM, 256), 256, 0, stream>>>(root2, basis2, Wp2);

    // ---- Phase A: concept layer on g1 ----
    zero_f32<<<ZB, 256, 0, stream>>>(deg, (size_t)N1);
    zero_f32<<<ZB, 256, 0, stream>>>(aggA, (size_t)N1 * DIM);
    count_deg<<<cdiv(E1, 256), 256, 0, stream>>>(dst_g1, deg, E1);
    concept_scatter<<<cdiv((long)E1 * 32, 256), 256, 0, stream>>>(x_in, src_g1, dst_g1, aggA, E1);
    concept_finalize<<<ZB, 256, 0, stream>>>(x_in, aggA, deg, xg1, (size_t)N1 * DIM);

    // ---- Relation counts on g2 (shared by both RGCN layers) ----
    zero_f32<<<ZB, 256, 0, stream>>>(cnt, (size_t)N2 * NREL);
    count_rel<<<cdiv(E2, 256), 256, 0, stream>>>(dst_g2, et_g2, cnt, E2);

    // ---- RGCN layer 1 ----
    zero_f32<<<ZB, 256, 0, stream>>>(aggb, (size_t)NB * N2 * DIM);
    rgcn_edge_scatter<<<cdiv((long)E2 * 32, 256), 256, 0, stream>>>(
        xg1, src_g2, dst_g2, et_g2, comp1, cnt, aggb, E2);
    rgcn_gemm_wmma<<<N2 / 16, 256, 0, stream>>>(xg1, aggb, Wp1, bias1, h1, 1);

    // ---- RGCN layer 2 ----
    zero_f32<<<ZB, 256, 0, stream>>>(aggb, (size_t)NB * N2 * DIM);
    rgcn_edge_scatter<<<cdiv((long)E2 * 32, 256), 256, 0, stream>>>(
        h1, src_g2, dst_g2, et_g2, comp2, cnt, aggb, E2);
    rgcn_gemm_wmma<<<N2 / 16, 256, 0, stream>>>(h1, aggb, Wp2, bias2, h2, 0);

    // ---- Logits + softmax ----
    make_tyw<<<cdiv(LCOM * DIM, 256), 256, 0, stream>>>(xg1, lc, wvec, tyw);
    logits_wmma<<<N2 / 16, 256, 0, stream>>>(h2, tyw, logits);
    softmax_rows<<<cdiv(N2, 8), 256, 0, stream>>>(logits, N2);
}